// MoERouter_16887811408648
// MI455X (gfx1250) — compile-verified
//
#include <hip/hip_runtime.h>
#include <hip/hip_bf16.h>

typedef __attribute__((ext_vector_type(2))) float v2f;
typedef __attribute__((ext_vector_type(8))) float v8f;

#define D_DIM 2048
#define E_EXP 64
#define K_ACT 8
#define KC 64            // K-chunk staged in LDS
#define MT 64            // tokens per block
#define XS_STRIDE 68     // 64 + 4 pad: de-conflicts row-strided b64 LDS reads
#define AFF_STRIDE 65    // 64 + 1 pad: de-conflicts per-token row scans
#define ALPHA 1e-4f
#define EPS 1e-9f

__global__ __launch_bounds__(128) void moe_router_kernel(
    const float* __restrict__ x, const float* __restrict__ Wg,
    const float* __restrict__ bias, float* __restrict__ out,
    float* __restrict__ gCnt, float* __restrict__ gP, int T)
{
    __shared__ float xs [2][MT * XS_STRIDE];    // double-buffered x tile (64 tok x 64 k)
    __shared__ float wsh[2][E_EXP * XS_STRIDE]; // double-buffered W tile (64 exp x 64 k)
    __shared__ float aff[MT * AFF_STRIDE];      // sigmoid logits (64 tok x 64 exp)
    __shared__ float rowsum[MT];
    __shared__ float biasS[E_EXP];
    __shared__ int   cnt[E_EXP];

    const int tid  = threadIdx.x;
    const int wave = tid >> 5;       // 0..3 -> expert group
    const int lane = tid & 31;
    const int tok0 = blockIdx.x * MT;

    if (tid < E_EXP) { biasS[tid] = bias[tid]; cnt[tid] = 0; }

    // LDS byte offsets (flat LDS aperture: low 32 bits = wave-relative offset)
    const unsigned xsb[2]  = { (unsigned)(uintptr_t)&xs[0][0],
                               (unsigned)(uintptr_t)&xs[1][0] };
    const unsigned wshb[2] = { (unsigned)(uintptr_t)&wsh[0][0],
                               (unsigned)(uintptr_t)&wsh[1][0] };

    // Per-thread slice of the cooperative tile fill: 1024 b128 slots per tile,
    // thread covers (row = tid>>4 + 8*it, col4 = tid&15) — coalesced, 16B aligned.
    const int r0 = tid >> 4;
    const int c4 = (tid & 15) << 2;

    // issue async fills (memory -> LDS, no VGPR staging, tracked by ASYNCcnt)
    auto issue_chunk = [&](int buf, int kk) {
        unsigned lds_x = xsb[buf]  + (unsigned)((r0 * XS_STRIDE + c4) << 2);
        unsigned lds_w = wshb[buf] + (unsigned)((r0 * XS_STRIDE + c4) << 2);
        unsigned vox = (unsigned)(((tok0 + r0) * D_DIM + c4) << 2);  // byte offset in x
        unsigned vow = (unsigned)((r0 * D_DIM + c4) << 2);           // byte offset in W
        const float* sx = x  + kk;   // uniform SGPR base (GVS mode)
        const float* sw = Wg + kk;
        #pragma unroll
        for (int it = 0; it < 8; ++it) {
            asm volatile("global_load_async_to_lds_b128 %0, %1, %2"
                         :: "v"(lds_x), "v"(vox), "s"(sx) : "memory");
            asm volatile("global_load_async_to_lds_b128 %0, %1, %2"
                         :: "v"(lds_w), "v"(vow), "s"(sw) : "memory");
            lds_x += 8u * XS_STRIDE * 4u;  lds_w += 8u * XS_STRIDE * 4u;
            vox   += 8u * D_DIM * 4u;      vow   += 8u * D_DIM * 4u;
        }
    };

    v8f acc0 = {}, acc1 = {}, acc2 = {}, acc3 = {};

    const int m16 = lane & 15;             // row within 16x16 fragment
    const int khi = (lane >> 4) << 1;      // +2 k for high half-wave (f32 16x16x4 layout)

    // prologue: fill buffer 0
    issue_chunk(0, 0);
    asm volatile("s_wait_asynccnt 0x0" ::: "memory");
    __syncthreads();

    int cur = 0;
    for (int kk = 0; kk < D_DIM; kk += KC) {
        if (kk + KC < D_DIM) issue_chunk(cur ^ 1, kk + KC);  // overlap next fill

        const float* xb = &xs[cur][0];
        const float* wb = &wsh[cur][0];
        #pragma unroll
        for (int k4 = 0; k4 < KC / 4; ++k4) {
            const int kb = (k4 << 2) + khi;
            // B fragment: 4x16 slice of W^T for this wave's 16 experts
            v2f b  = *(const v2f*)&wb[(wave * 16 + m16) * XS_STRIDE + kb];
            // A fragments: four 16x4 token tiles (B reused 4x)
            v2f a0 = *(const v2f*)&xb[( 0 + m16) * XS_STRIDE + kb];
            v2f a1 = *(const v2f*)&xb[(16 + m16) * XS_STRIDE + kb];
            v2f a2 = *(const v2f*)&xb[(32 + m16) * XS_STRIDE + kb];
            v2f a3 = *(const v2f*)&xb[(48 + m16) * XS_STRIDE + kb];
            acc0 = __builtin_amdgcn_wmma_f32_16x16x4_f32(false, a0, false, b, (short)0, acc0, false, false);
            acc1 = __builtin_amdgcn_wmma_f32_16x16x4_f32(false, a1, false, b, (short)0, acc1, false, false);
            acc2 = __builtin_amdgcn_wmma_f32_16x16x4_f32(false, a2, false, b, (short)0, acc2, false, false);
            acc3 = __builtin_amdgcn_wmma_f32_16x16x4_f32(false, a3, false, b, (short)0, acc3, false, false);
        }

        // own async fills done, then all-waves: fills visible + cur fully consumed
        asm volatile("s_wait_asynccnt 0x0" ::: "memory");
        __syncthreads();
        cur ^= 1;
    }

    // scatter C tiles -> aff with sigmoid applied.
    // C layout: VGPR r, lanes 0-15: M=r, lanes 16-31: M=r+8; N = lane&15
    {
        const int base_n = wave * 16 + m16;
        const int hi = (lane >> 4) << 3;
        #pragma unroll
        for (int r = 0; r < 8; ++r) {
            aff[( 0 + r + hi) * AFF_STRIDE + base_n] = 1.f / (1.f + __expf(-acc0[r]));
            aff[(16 + r + hi) * AFF_STRIDE + base_n] = 1.f / (1.f + __expf(-acc1[r]));
            aff[(32 + r + hi) * AFF_STRIDE + base_n] = 1.f / (1.f + __expf(-acc2[r]));
            aff[(48 + r + hi) * AFF_STRIDE + base_n] = 1.f / (1.f + __expf(-acc3[r]));
        }
    }
    __syncthreads();

    // one thread per token: row-sum + stable top-8 (selection by aff+bias, gate by aff)
    if (tid < MT) {
        const int trow = tid * AFF_STRIDE;
        float rv[K_ACT], gv[K_ACT];
        int   gi[K_ACT];
        #pragma unroll
        for (int j = 0; j < K_ACT; ++j) { rv[j] = -3.4e38f; gv[j] = 0.f; gi[j] = 0; }
        float s = 0.f;
        for (int e = 0; e < E_EXP; ++e) {
            float a = aff[trow + e];
            s += a;
            float r = a + biasS[e];
            if (r > rv[K_ACT - 1]) {             // strict >: lowest index wins ties (JAX top_k)
                int p = K_ACT - 1;
                while (p > 0 && r > rv[p - 1]) {
                    rv[p] = rv[p - 1]; gv[p] = gv[p - 1]; gi[p] = gi[p - 1]; --p;
                }
                rv[p] = r; gv[p] = a; gi[p] = e;
            }
        }
        rowsum[tid] = s;
        float gsum = 0.f;
        #pragma unroll
        for (int j = 0; j < K_ACT; ++j) gsum += gv[j];
        const float inv = 1.f / (gsum + EPS);
        const size_t tok = (size_t)tok0 + tid;
        #pragma unroll
        for (int j = 0; j < K_ACT; ++j) {
            out[tok * K_ACT + j] = gv[j] * inv;
            out[(size_t)T * K_ACT + tok * K_ACT + j] = (float)gi[j];  // indices as float
            atomicAdd(&cnt[gi[j]], 1);
        }
    }
    __syncthreads();

    // per-expert partials: P (normalized affinity) and selection counts -> global
    if (tid < E_EXP) {
        float p = 0.f;
        for (int t = 0; t < MT; ++t)
            p += aff[t * AFF_STRIDE + tid] / (rowsum[t] + EPS);
        atomicAdd(&gP[tid], p);
        atomicAdd(&gCnt[tid], (float)cnt[tid]);
    }
}

__global__ void moe_loss_kernel(const float* __restrict__ gCnt,
                                const float* __restrict__ gP,
                                float* __restrict__ out, int T)
{
    if (threadIdx.x == 0) {
        const float fscale = (float)E_EXP / ((float)K_ACT * (float)T);
        float loss = 0.f;
        for (int e = 0; e < E_EXP; ++e)
            loss += (gCnt[e] * fscale) * (gP[e] / (float)T);
        out[(size_t)2 * (size_t)T * K_ACT] = ALPHA * loss;
    }
}

extern "C" void kernel_launch(void* const* d_in, const int* in_sizes, int n_in,
                              void* d_out, int out_size, void* d_ws, size_t ws_size,
                              hipStream_t stream) {
    const float* x    = (const float*)d_in[0];
    const float* Wg   = (const float*)d_in[1];
    const float* bias = (const float*)d_in[2];
    float* out = (float*)d_out;
    const int T = in_sizes[0] / D_DIM;   // 16384

    float* gCnt = (float*)d_ws;
    float* gP   = gCnt + E_EXP;
    hipMemsetAsync(d_ws, 0, 2 * E_EXP * sizeof(float), stream);

    moe_router_kernel<<<T / MT, 128, 0, stream>>>(x, Wg, bias, out, gCnt, gP, T);
    moe_loss_kernel<<<1, 32, 0, stream>>>(gCnt, gP, out, T);
}